// CriteriaRelationNetwork_41867341201760
// MI455X (gfx1250) — compile-verified
//
#include <hip/hip_runtime.h>

#define NNODES 50000
#define NEDGES 800000
#define FIN   128
#define FHID  384
#define FOUT  256

typedef __attribute__((ext_vector_type(2))) float v2f;
typedef __attribute__((ext_vector_type(8))) float v8f;

// Guaranteed hardware atomic: L2-side f32 add, device scope, no return (STOREcnt).
__device__ __forceinline__ void atomic_add_f32(float* p, float v) {
    asm volatile("global_atomic_add_f32 %0, %1, off scope:SCOPE_DEV"
                 :: "v"(p), "v"(v)
                 : "memory");
}

// Async copy 16B from global to LDS (ASYNCcnt-tracked, no VGPR round-trip).
__device__ __forceinline__ void async_copy_b128(void* lds_ptr, const float* gptr) {
    unsigned lds_off = (unsigned)(uintptr_t)lds_ptr;   // generic LDS addr: low 32 bits = wave-relative offset
    asm volatile("global_load_async_to_lds_b128 %0, %1, off"
                 :: "v"(lds_off), "v"(gptr)
                 : "memory");
}

__device__ __forceinline__ void wait_async_zero() {
    asm volatile("s_wait_asynccnt 0" ::: "memory");
}

// ---------------- zero ----------------
__global__ void zero_kernel(float4* __restrict__ p, int n4) {
    int i = blockIdx.x * blockDim.x + threadIdx.x;
    int stride = gridDim.x * blockDim.x;
    for (; i < n4; i += stride) p[i] = make_float4(0.f, 0.f, 0.f, 0.f);
}

// ---------------- degrees ----------------
__global__ void degree_kernel(const int* __restrict__ src, const int* __restrict__ dst,
                              float* __restrict__ deg_out, float* __restrict__ deg_in) {
    int e = blockIdx.x * blockDim.x + threadIdx.x;
    if (e < NEDGES) {
        atomic_add_f32(&deg_out[src[e]], 1.0f);
        atomic_add_f32(&deg_in[dst[e]], 1.0f);
    }
}

// ---------------- norms ----------------
__global__ void norm_kernel(const float* __restrict__ deg_out, const float* __restrict__ deg_in,
                            float* __restrict__ out_norm, float* __restrict__ in_norm) {
    int i = blockIdx.x * blockDim.x + threadIdx.x;
    if (i < NNODES) {
        out_norm[i] = 1.0f / sqrtf(fmaxf(deg_out[i], 1.0f));
        in_norm[i]  = 1.0f / sqrtf(fmaxf(deg_in[i], 1.0f));
    }
}

// ---------------- rowwise scale (features * out_norm) ----------------
template<int COLS>
__global__ void scale_rows_kernel(const float4* __restrict__ in, const float* __restrict__ scale,
                                  float4* __restrict__ out, int n4) {
    int i = blockIdx.x * blockDim.x + threadIdx.x;
    if (i >= n4) return;
    int row = (i * 4) / COLS;
    float s = scale[row];
    float4 v = in[i];
    v.x *= s; v.y *= s; v.z *= s; v.w *= s;
    out[i] = v;
}

// ---------------- edge scatter-add: one wave per edge ----------------
template<int FEAT>
__global__ __launch_bounds__(256) void scatter_add_kernel(const int* __restrict__ src,
                                                          const int* __restrict__ dst,
                                                          const float* __restrict__ msg,
                                                          float* __restrict__ acc) {
    int wave = threadIdx.x >> 5;
    int lane = threadIdx.x & 31;
    int e = blockIdx.x * 8 + wave;
    if (e >= NEDGES) return;
    int s = src[e];
    int d = dst[e];
    const float4* mrow = (const float4*)(msg + (size_t)s * FEAT);
    float* arow = acc + (size_t)d * FEAT;
#pragma unroll
    for (int i = 0; i < FEAT / 128; ++i) {
        int q = lane + i * 32;          // float4 index within row
        float4 v = mrow[q];
        int c = q * 4;
        atomic_add_f32(&arow[c + 0], v.x);
        atomic_add_f32(&arow[c + 1], v.y);
        atomic_add_f32(&arow[c + 2], v.z);
        atomic_add_f32(&arow[c + 3], v.w);
    }
}

// ---------------- WMMA f32 GEMM: D = epilogue(rowscale[r] * (A @ B)) ----------------
// A: [M, KDIM] row-major, B: [KDIM, NDIM] row-major, D: [M, NDIM].
// Block: 256 thr = 8 waves; each wave one 16x16 tile; block tile = 16 rows x 128 cols.
// grid = (NDIM/128, M/16). M=50000 (=3125*16), NDIM in {384,256} -> exact tiling.
// Row scaling (in_norm for GEMM1, out_norm for GEMM2) is applied in the epilogue
// BEFORE bias+relu: diag(s)*(A@B) == (diag(s)*A)@B, so this matches the reference.
template<int KDIM, int NDIM, bool BIAS_RELU>
__global__ __launch_bounds__(256) void gemm_wmma_kernel(const float* __restrict__ A,
                                                        const float* __restrict__ B,
                                                        const float* __restrict__ bias,
                                                        const float* __restrict__ rowscale,
                                                        float* __restrict__ D) {
    __shared__ float As[16][KDIM + 4];   // +4 pad: stride%64==4 banks -> conflict-free

    const int tid  = threadIdx.x;
    const int lane = tid & 31;
    const int wave = tid >> 5;
    const int hi   = lane >> 4;          // which half of the wave
    const int l16  = lane & 15;
    const int row0 = blockIdx.y * 16;
    const int col  = blockIdx.x * 128 + wave * 16 + l16;

    // async-stage A tile (16 x KDIM) straight into LDS
    {
        constexpr int PER = (16 * KDIM / 4) / 256;   // float4s per thread: 2 (K=128) or 6 (K=384)
#pragma unroll
        for (int t = 0; t < PER; ++t) {
            int f   = tid + t * 256;            // coalesced float4 index
            int row = (f * 4) / KDIM;
            int c   = (f * 4) % KDIM;
            async_copy_b128(&As[row][c], A + (size_t)(row0 + row) * KDIM + c);
        }
        wait_async_zero();
    }
    __syncthreads();

    v8f acc = {};
#pragma unroll 8
    for (int k0 = 0; k0 < KDIM; k0 += 4) {
        const int kk = k0 + 2 * hi;
        // A frag 16x4: lanes 0-15 hold K={k0,k0+1}, lanes 16-31 K={k0+2,k0+3}
        v2f a = *(const v2f*)&As[l16][kk];
        // B frag 4x16: same K split across wave halves, N = l16 within half
        v2f b;
        b.x = B[(size_t)kk * NDIM + col];
        b.y = B[(size_t)(kk + 1) * NDIM + col];
        acc = __builtin_amdgcn_wmma_f32_16x16x4_f32(false, a, false, b,
                                                    (short)0, acc, false, false);
    }

    const float bv = BIAS_RELU ? bias[col] : 0.0f;
#pragma unroll
    for (int r = 0; r < 8; ++r) {
        // C/D layout: VGPR r, lanes 0-15 -> M=r, lanes 16-31 -> M=r+8
        const int row = row0 + r + 8 * hi;
        float v = acc[r] * rowscale[row];
        if (BIAS_RELU) v = fmaxf(v + bv, 0.0f);
        D[(size_t)row * NDIM + col] = v;
    }
}

// ---------------- finalize: out = out * in_norm[row] + b2[col] ----------------
__global__ void finalize_kernel(float4* __restrict__ out, const float* __restrict__ in_norm,
                                const float* __restrict__ b2) {
    int i = blockIdx.x * blockDim.x + threadIdx.x;
    const int n4 = NNODES * FOUT / 4;
    if (i >= n4) return;
    int j = i * 4;
    int row = j >> 8;       // / 256
    int c   = j & 255;
    float s = in_norm[row];
    float4 v = out[i];
    float4 b = *(const float4*)(b2 + c);
    v.x = v.x * s + b.x;
    v.y = v.y * s + b.y;
    v.z = v.z * s + b.z;
    v.w = v.w * s + b.w;
    out[i] = v;
}

extern "C" void kernel_launch(void* const* d_in, const int* in_sizes, int n_in,
                              void* d_out, int out_size, void* d_ws, size_t ws_size,
                              hipStream_t stream) {
    (void)in_sizes; (void)n_in; (void)out_size; (void)ws_size;
    const float* features = (const float*)d_in[0];
    const float* W1       = (const float*)d_in[1];
    const float* b1       = (const float*)d_in[2];
    const float* W2       = (const float*)d_in[3];
    const float* b2       = (const float*)d_in[4];
    const int*   edges    = (const int*)d_in[5];
    const int*   src      = edges;
    const int*   dst      = edges + NEDGES;
    float*       out      = (float*)d_out;

    float* ws       = (float*)d_ws;
    float* deg_out  = ws;
    float* deg_in   = ws + NNODES;
    float* out_norm = ws + 2 * (size_t)NNODES;
    float* in_norm  = ws + 3 * (size_t)NNODES;
    float* bufA     = ws + 4 * (size_t)NNODES;            // N*384 floats
    float* bufB     = bufA + (size_t)NNODES * FHID;       // N*256 floats
    float* scaled_feat = bufA;   // [N,128]
    float* agg1        = bufB;   // [N,128]
    float* h1          = bufA;   // [N,384] (scaled_feat dead by then)
    float* t2s         = bufB;   // [N,256] (agg1 dead by then)

    const int B = 256;

    // degrees + norms
    zero_kernel<<<(2 * NNODES / 4 + B - 1) / B, B, 0, stream>>>((float4*)deg_out, 2 * NNODES / 4);
    degree_kernel<<<(NEDGES + B - 1) / B, B, 0, stream>>>(src, dst, deg_out, deg_in);
    norm_kernel<<<(NNODES + B - 1) / B, B, 0, stream>>>(deg_out, deg_in, out_norm, in_norm);

    // conv1 aggregate: scaled_feat = features*out_norm ; agg1 = scatter_add
    {
        const int n4 = NNODES * FIN / 4;
        scale_rows_kernel<FIN><<<(n4 + B - 1) / B, B, 0, stream>>>(
            (const float4*)features, out_norm, (float4*)scaled_feat, n4);
        zero_kernel<<<(n4 + B - 1) / B, B, 0, stream>>>((float4*)agg1, n4);
        scatter_add_kernel<FIN><<<NEDGES / 8, B, 0, stream>>>(src, dst, scaled_feat, agg1);
    }

    // conv1 dense: h1 = relu(in_norm[row]*(agg1 @ W1) + b1)
    gemm_wmma_kernel<FIN, FHID, true>
        <<<dim3(FHID / 128, NNODES / 16), B, 0, stream>>>(agg1, W1, b1, in_norm, h1);

    // conv2 dense: t2s = out_norm[row]*(h1 @ W2)
    gemm_wmma_kernel<FHID, FOUT, false>
        <<<dim3(FOUT / 128, NNODES / 16), B, 0, stream>>>(h1, W2, nullptr, out_norm, t2s);

    // conv2 aggregate into out, then finalize with in_norm and b2
    {
        const int n4 = NNODES * FOUT / 4;
        zero_kernel<<<(n4 + B - 1) / B, B, 0, stream>>>((float4*)out, n4);
        scatter_add_kernel<FOUT><<<NEDGES / 8, B, 0, stream>>>(src, dst, t2s, out);
        finalize_kernel<<<(n4 + B - 1) / B, B, 0, stream>>>((float4*)out, in_norm, b2);
    }
}